// OptNetIneq_41970420417260
// MI455X (gfx1250) — compile-verified
//
#include <hip/hip_runtime.h>

// OptNetIneq batched primal-dual IPM for MI455X (gfx1250).
// One workgroup (256 thr = 8 wave32) per batch item; full 25-iteration IPM in
// LDS. Linear solve uses Woodbury: K = D + G2^T D2 G2 (rank-128 update):
//   S = D2^-1 + G2 D^-1 G2^T   (128x128, formed with V_WMMA_F32_16X16X4_F32)
// then a BLOCKED Cholesky of S (16-wide panels): diagonal blocks factored
// wave-synchronously by wave 0, trailing rank-16 SYRK updates via WMMA.
// G2 is transposed once into d_ws (G2t[729][128]) so WMMA fragment loads and
// row-matvecs are coalesced (row-major G2 columns are stride-2916B scattered).

#define NX    729
#define ML    128
#define MT    857          // NX + ML
#define NT    256
#define SP    129          // S row stride (129 % 64 == 1 -> conflict-free cols)
#define NITER 25
#define QPENF 0.1f
#define REGF  1e-6f
#define SIGF  0.1f

typedef __attribute__((ext_vector_type(2))) float v2f;
typedef __attribute__((ext_vector_type(8))) float v8f;

// Total dynamic LDS floats:
//  S 128*129 + 5*729 (z,p,invD,w,dz) + 2*857 (lam,s) + 6*128 (h2,rp2,t,um,ds2,mv) + 256 red
#define SMEM_FLOATS (128*SP + 5*NX + 2*MT + 6*ML + NT)

// ---- wave32 shuffle reductions (no barriers), combined across 8 waves ----
__device__ __forceinline__ float wave_sum(float v) {
  #pragma unroll
  for (int m = 16; m >= 1; m >>= 1) v += __shfl_xor(v, m, 32);
  return v;
}
__device__ __forceinline__ float wave_min(float v) {
  #pragma unroll
  for (int m = 16; m >= 1; m >>= 1) v = fminf(v, __shfl_xor(v, m, 32));
  return v;
}
__device__ __forceinline__ float blk_sum(float v, float* red, int tid) {
  v = wave_sum(v);
  if ((tid & 31) == 0) red[tid >> 5] = v;
  __syncthreads();
  float r = ((red[0] + red[1]) + (red[2] + red[3])) +
            ((red[4] + red[5]) + (red[6] + red[7]));
  __syncthreads();
  return r;
}
__device__ __forceinline__ float blk_min(float v, float* red, int tid) {
  v = wave_min(v);
  if ((tid & 31) == 0) red[tid >> 5] = v;
  __syncthreads();
  float r = fminf(fminf(fminf(red[0], red[1]), fminf(red[2], red[3])),
                  fminf(fminf(red[4], red[5]), fminf(red[6], red[7])));
  __syncthreads();
  return r;
}

// out[i] = (G2 @ x)[i] via row-major G2 (fallback; column-strided reads).
__device__ __forceinline__ void g2_matvec(const float* __restrict__ G2,
                                          const float* __restrict__ x,
                                          float* out, float* red, int tid) {
  const int i  = tid & (ML - 1);
  const int hf = tid >> 7;
  const int k0 = hf ? 365 : 0;
  const int k1 = hf ? NX  : 365;
  const float* row = G2 + i * NX;
  float acc = 0.f;
  #pragma unroll 8
  for (int k = k0; k < k1; ++k) acc = fmaf(row[k], x[k], acc);
  red[tid] = acc; __syncthreads();
  if (tid < ML) out[tid] = red[tid] + red[tid + ML];
  __syncthreads();
}

// out[i] = (G2 @ x)[i] via transposed G2t[729][128]: fully coalesced loads.
__device__ __forceinline__ void g2_matvec_t(const float* __restrict__ G2t,
                                            const float* __restrict__ x,
                                            float* out, float* red, int tid) {
  const int i  = tid & (ML - 1);
  const int hf = tid >> 7;
  const int k0 = hf ? 365 : 0;
  const int k1 = hf ? NX  : 365;
  float acc = 0.f;
  #pragma unroll 4
  for (int k = k0; k < k1; ++k) acc = fmaf(G2t[(size_t)k * ML + i], x[k], acc);
  red[tid] = acc; __syncthreads();
  if (tid < ML) out[tid] = red[tid] + red[tid + ML];
  __syncthreads();
}

__launch_bounds__(NT, 1)
__global__ void optnet_ipm_kernel(const float* __restrict__ puzzles,
                                  const float* __restrict__ G2,
                                  const float* __restrict__ z2,
                                  const float* __restrict__ s2,
                                  float* __restrict__ out,
                                  float* __restrict__ G2t) {
  extern __shared__ float sm[];
  float* S    = sm;                 // 128*SP
  float* zv   = S    + 128 * SP;    // 729
  float* pv   = zv   + NX;          // 729  (p = -puzzle)
  float* invD = pv   + NX;          // 729
  float* wv   = invD + NX;          // 729  (rhs -> w = D^-1 rhs)
  float* dzv  = wv   + NX;          // 729
  float* lamv = dzv  + NX;          // 857
  float* sv   = lamv + MT;          // 857
  float* h2v  = sv   + MT;          // 128
  float* rp2v = h2v  + ML;          // 128
  float* tv   = rp2v + ML;          // 128  (t -> y after solve)
  float* umv  = tv   + ML;          // 128  (u2 - lam2)
  float* ds2v = umv  + ML;          // 128
  float* mvb  = ds2v + ML;          // 128  (matvec scratch)
  float* red  = mvb  + ML;          // 256

  const int tid = threadIdx.x;
  const int b   = blockIdx.x;
  const bool use_t = (G2t != nullptr);
  const float* pz = puzzles + (size_t)b * NX;

  // ---- one-time tiled transpose G2[128][729] -> G2t[729][128] ----
  // Every block writes identical values (benign, deterministic); each block
  // only reads what it wrote itself after the barrier. 64x64 tiles via LDS
  // (staging tile aliases S, which is unused until the first iteration).
  if (use_t) {
    float* T = S;                          // 64*65 = 4160 <= 128*SP
    for (int R = 0; R < ML; R += 64) {
      for (int C = 0; C < NX; C += 64) {
        __syncthreads();
        {                                   // coalesced row reads
          const int c = tid & 63, r0 = tid >> 6;
          for (int r = r0; r < 64; r += 4) {
            const int gc = C + c;
            T[r * 65 + c] = (gc < NX) ? G2[(size_t)(R + r) * NX + gc] : 0.f;
          }
        }
        __syncthreads();
        {                                   // coalesced column writes
          const int r2 = tid & 63, c20 = tid >> 6;
          for (int c2 = c20; c2 < 64; c2 += 4) {
            const int gc = C + c2;
            if (gc < NX) G2t[(size_t)gc * ML + R + r2] = T[r2 * 65 + c2];
          }
        }
      }
    }
    __syncthreads();
  }

  // ---- init: p = -puzzle, z = -p/Q, lam = 1, s1 = max(z, 1) ----
  for (int k = tid; k < NX; k += NT) {
    float pu = pz[k];
    pv[k]   = -pu;
    float zk = pu / QPENF;
    zv[k]   = zk;
    lamv[k] = 1.f;
    sv[k]   = fmaxf(zk, 1.f);
  }
  __syncthreads();
  // h2 = G2 @ z2 + s2
  g2_matvec(G2, z2, h2v, red, tid);        // z2 read from global directly
  if (tid < ML) h2v[tid] += s2[tid];
  __syncthreads();
  // s2part = max(h2 - G2 z, 1), lam2 = 1
  if (use_t) g2_matvec_t(G2t, zv, mvb, red, tid);
  else       g2_matvec(G2, zv, mvb, red, tid);
  if (tid < ML) {
    sv[NX + tid]   = fmaxf(h2v[tid] - mvb[tid], 1.f);
    lamv[NX + tid] = 1.f;
  }
  __syncthreads();

  // ---- WMMA lane geometry (fp32 16x16x4, ISA 7.12.2 layouts) ----
  const int ln     = tid & 31;        // lane in wave32
  const int wid    = tid >> 5;        // wave id 0..7
  const int mrow   = ln & 15;         // M (A) / N (B) index
  const int kh     = (ln >> 4) << 1;  // K sub-offset: 0 or 2
  const int rowoff = (ln >> 4) * 8;   // C/D: row = v + 8*(lane>=16)
  const int rbase  = wid * 16;
  const int colA   = rbase + mrow;    // G2t column for this wave's A rows
  const float* Arow = G2 + (size_t)colA * NX;   // fallback path

  for (int it = 0; it < NITER; ++it) {
    // rp2 = G2 z + s2part - h2
    if (use_t) g2_matvec_t(G2t, zv, mvb, red, tid);
    else       g2_matvec(G2, zv, mvb, red, tid);
    if (tid < ML) rp2v[tid] = mvb[tid] + sv[NX + tid] - h2v[tid];
    __syncthreads();

    // mu = SIGMA * dot(lam, s) / m
    float loc = 0.f;
    for (int j = tid; j < MT; j += NT) loc = fmaf(lamv[j], sv[j], loc);
    const float mu = SIGF * blk_sum(loc, red, tid) / (float)MT;

    // um = u2 - lam2, with u2 = (rc2 - lam2*rp2)/s2
    if (tid < ML) {
      const int gi = NX + tid;
      float l2 = lamv[gi], s2e = sv[gi];
      float rc2 = l2 * s2e - mu;
      float u2  = (rc2 - l2 * rp2v[tid]) / s2e;
      umv[tid]  = u2 - l2;
    }
    // invD = 1 / (QPEN + REG + lam1/s1)
    for (int k = tid; k < NX; k += NT)
      invD[k] = 1.f / (QPENF + REGF + lamv[k] / sv[k]);
    __syncthreads();

    // rhs = -QPEN z - p + lam1 - u1 + G2^T (u2 - lam2);  w = rhs * invD
    // (row-major G2: consecutive k-threads read consecutive addresses)
    for (int k = tid; k < NX; k += NT) {
      float g = 0.f;
      #pragma unroll 8
      for (int i = 0; i < ML; ++i) g = fmaf(G2[(size_t)i * NX + k], umv[i], g);
      float zk = zv[k], l1 = lamv[k], s1 = sv[k];
      float rp1 = s1 - zk;
      float rc1 = l1 * s1 - mu;
      float u1  = (rc1 - l1 * rp1) / s1;
      float rhs = -(QPENF * zk + pv[k]) + l1 - u1 + g;
      wv[k] = rhs * invD[k];
    }
    __syncthreads();

    // t = G2 @ w
    if (use_t) g2_matvec_t(G2t, wv, tv, red, tid);
    else       g2_matvec(G2, wv, tv, red, tid);

    // ---- S = G2 diag(invD) G2^T via V_WMMA_F32_16X16X4_F32 ----
    {
      v8f acc[8];
      #pragma unroll
      for (int tj = 0; tj < 8; ++tj) acc[tj] = (v8f){0,0,0,0,0,0,0,0};
      if (use_t) {
        // G2t path: for fixed k, lanes 0-15 read consecutive floats.
        for (int kk = 0; kk < 728; kk += 4) {
          const int k0 = kk + kh;
          const float* r0 = G2t + (size_t)k0 * ML;
          const float* r1 = r0 + ML;
          v2f Af = { r0[colA] * invD[k0], r1[colA] * invD[k0 + 1] };
          #pragma unroll
          for (int tj = 0; tj < 8; ++tj) {
            const int cB = tj * 16 + mrow;
            v2f Bf = { r0[cB], r1[cB] };
            acc[tj] = __builtin_amdgcn_wmma_f32_16x16x4_f32(
                false, Af, false, Bf, (short)0, acc[tj], false, false);
          }
        }
        { // tail kk = 728 (only k==728 valid, only kh==0 lanes)
          const int k0 = 728 + kh;
          const bool g0 = (k0 < NX);
          const float* r0 = G2t + (size_t)(g0 ? k0 : 0) * ML;
          v2f Af = { g0 ? r0[colA] * invD[k0] : 0.f, 0.f };
          #pragma unroll
          for (int tj = 0; tj < 8; ++tj) {
            v2f Bf = { g0 ? r0[tj * 16 + mrow] : 0.f, 0.f };
            acc[tj] = __builtin_amdgcn_wmma_f32_16x16x4_f32(
                false, Af, false, Bf, (short)0, acc[tj], false, false);
          }
        }
      } else {
        // fallback: row-major G2 (scattered but correct)
        for (int kk = 0; kk < 728; kk += 4) {
          const int k0 = kk + kh;
          v2f Af = { Arow[k0] * invD[k0], Arow[k0 + 1] * invD[k0 + 1] };
          #pragma unroll
          for (int tj = 0; tj < 8; ++tj) {
            const float* Brow = G2 + (size_t)(tj * 16 + mrow) * NX;
            v2f Bf = { Brow[k0], Brow[k0 + 1] };
            acc[tj] = __builtin_amdgcn_wmma_f32_16x16x4_f32(
                false, Af, false, Bf, (short)0, acc[tj], false, false);
          }
        }
        {
          const int k0 = 728 + kh;
          const bool g0 = (k0 < NX);
          v2f Af = { g0 ? Arow[k0] * invD[k0] : 0.f, 0.f };
          #pragma unroll
          for (int tj = 0; tj < 8; ++tj) {
            const float* Brow = G2 + (size_t)(tj * 16 + mrow) * NX;
            v2f Bf = { g0 ? Brow[k0] : 0.f, 0.f };
            acc[tj] = __builtin_amdgcn_wmma_f32_16x16x4_f32(
                false, Af, false, Bf, (short)0, acc[tj], false, false);
          }
        }
      }
      #pragma unroll
      for (int tj = 0; tj < 8; ++tj)
        #pragma unroll
        for (int v = 0; v < 8; ++v)
          S[(size_t)(rbase + rowoff + v) * SP + tj * 16 + mrow] = acc[tj][v];
    }
    __syncthreads();
    // S += diag(1/d2) = diag(s2/lam2)
    if (tid < ML) S[(size_t)tid * SP + tid] += sv[NX + tid] / lamv[NX + tid];
    __syncthreads();

    // ---- blocked Cholesky S = L L^T (16-wide panels, 8 blocks) ----
    for (int jb = 0; jb < 8; ++jb) {
      const int j0 = jb * 16;
      // (1) factor 16x16 diagonal block: wave 0, wave-synchronous
      if (wid == 0) {
        for (int j = 0; j < 16; ++j) {
          float ljj = sqrtf(fmaxf(S[(size_t)(j0 + j) * SP + j0 + j], 1e-30f));
          if (ln == j) S[(size_t)(j0 + j) * SP + j0 + j] = ljj;
          if (ln > j && ln < 16) {
            const size_t r = (size_t)(j0 + ln) * SP;
            float lrj = S[r + j0 + j] / ljj;
            S[r + j0 + j] = lrj;
            for (int c = j + 1; c <= ln; ++c)
              S[r + j0 + c] = fmaf(-lrj, S[(size_t)(j0 + c) * SP + j0 + j],
                                   S[r + j0 + c]);
          }
        }
      }
      __syncthreads();
      // (2) panel: rows i >= j0+16: L[i][j0:j0+16] = A[i][.] * L11^-T
      for (int i = j0 + 16 + tid; i < ML; i += NT) {
        const size_t r = (size_t)i * SP;
        #pragma unroll
        for (int c = 0; c < 16; ++c) {
          float v = S[r + j0 + c];
          for (int k2 = 0; k2 < c; ++k2)
            v = fmaf(-S[r + j0 + k2], S[(size_t)(j0 + c) * SP + j0 + k2], v);
          S[r + j0 + c] = v / S[(size_t)(j0 + c) * SP + j0 + c];
        }
      }
      __syncthreads();
      // (3) trailing SYRK: A22 -= L21 L21^T via WMMA (lower-triangle tiles)
      const int c0 = j0 + 16;
      const int nb = (ML - c0) >> 4;
      if (nb > 0) {
        const int ntiles = nb * (nb + 1) / 2;
        for (int t = wid; t < ntiles; t += 8) {   // wave-uniform -> EXEC full
          int ti = 0;
          while ((ti + 1) * (ti + 2) / 2 <= t) ++ti;
          const int tj = t - ti * (ti + 1) / 2;
          const size_t ra = (size_t)(c0 + ti * 16 + mrow) * SP;  // A rows
          const size_t rb = (size_t)(c0 + tj * 16 + mrow) * SP;  // B (=L21^T)
          v8f cf;
          #pragma unroll
          for (int v = 0; v < 8; ++v)
            cf[v] = S[(size_t)(c0 + ti * 16 + rowoff + v) * SP +
                      c0 + tj * 16 + mrow];
          #pragma unroll
          for (int kk = 0; kk < 16; kk += 4) {
            const int kx = j0 + kk + kh;
            v2f Af = { -S[ra + kx], -S[ra + kx + 1] };   // -L21 tile
            v2f Bf = {  S[rb + kx],  S[rb + kx + 1] };   //  L21^T tile
            cf = __builtin_amdgcn_wmma_f32_16x16x4_f32(
                false, Af, false, Bf, (short)0, cf, false, false);
          }
          #pragma unroll
          for (int v = 0; v < 8; ++v)
            S[(size_t)(c0 + ti * 16 + rowoff + v) * SP +
              c0 + tj * 16 + mrow] = cf[v];
        }
      }
      __syncthreads();
    }

    // ---- blocked solve S y = t (forward then backward), y overwrites tv ----
    for (int jb = 0; jb < 8; ++jb) {
      const int j0 = jb * 16;
      if (wid == 0) {          // 16-elem diagonal solve, wave-synchronous
        for (int j = 0; j < 16; ++j) {
          float yj = tv[j0 + j] / S[(size_t)(j0 + j) * SP + j0 + j];
          if (ln == 0) tv[j0 + j] = yj;
          if (ln > j && ln < 16)
            tv[j0 + ln] = fmaf(-S[(size_t)(j0 + ln) * SP + j0 + j], yj,
                               tv[j0 + ln]);
        }
      }
      __syncthreads();
      for (int i = j0 + 16 + tid; i < ML; i += NT) {
        float acc = tv[i];
        #pragma unroll
        for (int c = 0; c < 16; ++c)
          acc = fmaf(-S[(size_t)i * SP + j0 + c], tv[j0 + c], acc);
        tv[i] = acc;
      }
      __syncthreads();
    }
    for (int jb = 7; jb >= 0; --jb) {
      const int j0 = jb * 16;
      if (wid == 0) {          // L^T diagonal solve
        for (int j = 15; j >= 0; --j) {
          float xj = tv[j0 + j] / S[(size_t)(j0 + j) * SP + j0 + j];
          if (ln == 0) tv[j0 + j] = xj;
          if (ln < j)  // L^T[i][j] = L[j][i]
            tv[j0 + ln] = fmaf(-S[(size_t)(j0 + j) * SP + j0 + ln], xj,
                               tv[j0 + ln]);
        }
      }
      __syncthreads();
      for (int i = tid; i < j0; i += NT) {
        float acc = tv[i];
        #pragma unroll
        for (int c = 0; c < 16; ++c)
          acc = fmaf(-S[(size_t)(j0 + c) * SP + i], tv[j0 + c], acc);
        tv[i] = acc;
      }
      __syncthreads();
    }

    // dz = w - invD * (G2^T y)   (row-major G2: coalesced across k-threads)
    for (int k = tid; k < NX; k += NT) {
      float g = 0.f;
      #pragma unroll 8
      for (int i = 0; i < ML; ++i) g = fmaf(G2[(size_t)i * NX + k], tv[i], g);
      dzv[k] = wv[k] - invD[k] * g;
    }
    __syncthreads();

    // ds2 = -rp2 - G2 dz
    if (use_t) g2_matvec_t(G2t, dzv, mvb, red, tid);
    else       g2_matvec(G2, dzv, mvb, red, tid);
    if (tid < ML) ds2v[tid] = -rp2v[tid] - mvb[tid];
    __syncthreads();

    // ---- fraction-to-boundary: alpha = min(1, 0.99*min(a_s, a_l)) ----
    float amin = 1e30f;
    for (int k = tid; k < NX; k += NT) {
      float zk = zv[k], s1 = sv[k], l1 = lamv[k];
      float ds1 = zk - s1 + dzv[k];                 // -rp1 + dz
      if (ds1 < 0.f) amin = fminf(amin, -s1 / ds1);
      float rc1 = l1 * s1 - mu;
      float dl1 = -(rc1 + l1 * ds1) / s1;
      if (dl1 < 0.f) amin = fminf(amin, -l1 / dl1);
    }
    if (tid < ML) {
      const int gi = NX + tid;
      float l2 = lamv[gi], s2e = sv[gi], d2s = ds2v[tid];
      if (d2s < 0.f) amin = fminf(amin, -s2e / d2s);
      float rc2 = l2 * s2e - mu;
      float dl2 = -(rc2 + l2 * d2s) / s2e;
      if (dl2 < 0.f) amin = fminf(amin, -l2 / dl2);
    }
    const float alpha = fminf(1.f, 0.99f * blk_min(amin, red, tid));

    // ---- update (each thread owns disjoint elements; recompute ds/dlam) ----
    for (int k = tid; k < NX; k += NT) {
      float zk = zv[k], s1 = sv[k], l1 = lamv[k];
      float ds1 = zk - s1 + dzv[k];
      float rc1 = l1 * s1 - mu;
      float dl1 = -(rc1 + l1 * ds1) / s1;
      zv[k]   = zk + alpha * dzv[k];
      sv[k]   = s1 + alpha * ds1;
      lamv[k] = l1 + alpha * dl1;
    }
    if (tid < ML) {
      const int gi = NX + tid;
      float l2 = lamv[gi], s2e = sv[gi], d2s = ds2v[tid];
      float rc2 = l2 * s2e - mu;
      float dl2 = -(rc2 + l2 * d2s) / s2e;
      sv[gi]   = s2e + alpha * d2s;
      lamv[gi] = l2 + alpha * dl2;
    }
    __syncthreads();
  }

  // ---- emit z ----
  for (int k = tid; k < NX; k += NT) out[(size_t)b * NX + k] = zv[k];
}

extern "C" void kernel_launch(void* const* d_in, const int* in_sizes, int n_in,
                              void* d_out, int out_size, void* d_ws, size_t ws_size,
                              hipStream_t stream) {
  const float* puzzles = (const float*)d_in[0];   // [64,9,9,9] f32
  const float* G2      = (const float*)d_in[1];   // [128,729]  f32
  const float* z2      = (const float*)d_in[2];   // [729]      f32
  const float* s2      = (const float*)d_in[3];   // [128]      f32
  float* out = (float*)d_out;                     // [64,9,9,9] f32

  // transposed-G2 scratch (coalesced WMMA fragment loads); fallback if small
  float* G2t = (ws_size >= (size_t)NX * ML * sizeof(float)) ? (float*)d_ws
                                                            : nullptr;
  const int B = in_sizes[0] / NX;                 // 64
  const size_t smem = (size_t)SMEM_FLOATS * sizeof(float);  // ~90 KB dynamic LDS
  optnet_ipm_kernel<<<dim3(B), dim3(NT), smem, stream>>>(puzzles, G2, z2, s2,
                                                         out, G2t);
}